// CAPE_12086037971325
// MI455X (gfx1250) — compile-verified
//
#include <hip/hip_runtime.h>

typedef __attribute__((ext_vector_type(2))) float v2f;
typedef __attribute__((ext_vector_type(8))) float v8f;

#define BX 2
#define HX 12
#define NX 2048
#define CX 64
#define LX 2048
#define TILE_M 16
#define THREADS 512   // 16 waves (wave32)
#define CHUNK 64      // row elements per lane in scan phase

__global__ __launch_bounds__(THREADS, 1)
void cape_pos_attn_kernel(const float* __restrict__ query,   // (B,H,N,C)
                          const float* __restrict__ attn,    // (B,H,N,N)
                          const float* __restrict__ pos,     // (C,L)
                          float* __restrict__ out)           // (B,H,N,N)
{
    extern __shared__ float ldsE[];            // TILE_M x LX floats = 128 KB

    const int lane = threadIdx.x & 31;
    const int wv   = threadIdx.x >> 5;         // wave id 0..15
    const int half = lane >> 4;                // 0 | 1
    const int lm   = lane & 15;                // 0..15

    const int tilesPerBH = NX / TILE_M;        // 128
    const int bh = blockIdx.x / tilesPerBH;    // 0..23
    const int n0 = (blockIdx.x % tilesPerBH) * TILE_M;

    // ---------------- Phase A: E_tile(16x2048) = Q_tile(16x64) x pos(64x2048)
    // f32 WMMA 16x16x4; K=64 -> 16 WMMA per 16x16 output block.
    // A layout (16x4 f32): VGPR0 = K {0 | 2} for lane halves, VGPR1 = K {1 | 3}.
    // B layout (4x16 f32): mirrored; D: VGPR v -> row (v + 8*half), col = lm.
    const float* Qt = query + ((size_t)bh * NX + n0) * CX;

    for (int blk = 0; blk < 8; ++blk) {
        const int ncol = (wv * 8 + blk) * 16 + lm;   // output column in [0,2048)
        v8f acc = {};
        #pragma unroll
        for (int t = 0; t < 16; ++t) {
            const int k = t * 4 + half * 2;
            v2f a = *(const v2f*)(Qt + lm * CX + k);            // 8B load, contiguous K pair
            v2f b;
            b.x = pos[(size_t)k       * LX + ncol];
            b.y = pos[(size_t)(k + 1) * LX + ncol];
            acc = __builtin_amdgcn_wmma_f32_16x16x4_f32(
                      false, a, false, b, (short)0, acc, false, false);
        }
        #pragma unroll
        for (int v = 0; v < 8; ++v)
            ldsE[(size_t)(v + 8 * half) * LX + ncol] = acc[v];
    }
    __syncthreads();

    // ---------------- Phase B: gates, reverse cumsum, interpolated gather
    const int   row  = n0 + wv;                         // wave wv owns this row
    const float* arow = attn + ((size_t)bh * NX + row) * NX;
    float*       orow = out  + ((size_t)bh * NX + row) * NX;
    const float* Erow = ldsE + (size_t)wv * LX;

    float g[CHUNK];
    const int j0 = lane * CHUNK;                        // contiguous chunk per lane
    float lsum = 0.0f;
    #pragma unroll
    for (int i = 0; i < CHUNK; i += 4) {
        const float4 x = *(const float4*)(arow + j0 + i);
        // g = 1 - sigmoid(x) = 1/(1+exp(x))  -> v_exp_f32 path
        g[i+0] = 1.0f / (1.0f + __expf(x.x));
        g[i+1] = 1.0f / (1.0f + __expf(x.y));
        g[i+2] = 1.0f / (1.0f + __expf(x.z));
        g[i+3] = 1.0f / (1.0f + __expf(x.w));
        lsum  += g[i+0] + g[i+1] + g[i+2] + g[i+3];
    }

    // Inclusive suffix scan of lane sums across the wave32, then make exclusive.
    float s = lsum;
    #pragma unroll
    for (int d = 1; d < 32; d <<= 1) {
        const float t = __shfl_down(s, d, 32);
        if (lane + d < 32) s += t;
    }
    float run = s - lsum;   // sum over all elements in lanes > this lane

    // Walk the chunk backwards: run becomes the inclusive reverse cumsum P_j.
    #pragma unroll
    for (int i = CHUNK - 1; i >= 0; --i) {
        run += g[i];
        const float P  = fminf(run, (float)(LX - 1));
        const float fl = floorf(P);
        const float fr = P - fl;
        const int  ifl = (int)fl;
        const int  icl = (int)ceilf(P);
        const float ef = Erow[ifl];
        const float ec = Erow[icl];
        g[i] = fr * ec + (1.0f - fr) * ef;   // reuse registers for result
    }

    #pragma unroll
    for (int i = 0; i < CHUNK; i += 4) {
        float4 r; r.x = g[i]; r.y = g[i+1]; r.z = g[i+2]; r.w = g[i+3];
        *(float4*)(orow + j0 + i) = r;
    }
}

extern "C" void kernel_launch(void* const* d_in, const int* in_sizes, int n_in,
                              void* d_out, int out_size, void* d_ws, size_t ws_size,
                              hipStream_t stream) {
    (void)in_sizes; (void)n_in; (void)out_size; (void)d_ws; (void)ws_size;
    const float* query = (const float*)d_in[0];
    const float* attn  = (const float*)d_in[1];
    const float* pos   = (const float*)d_in[2];
    float*       out   = (float*)d_out;

    const size_t lds_bytes = (size_t)TILE_M * LX * sizeof(float);  // 131072
    (void)hipFuncSetAttribute((const void*)cape_pos_attn_kernel,
                              hipFuncAttributeMaxDynamicSharedMemorySize,
                              (int)lds_bytes);

    const int grid = BX * HX * (NX / TILE_M);   // 3072 workgroups
    cape_pos_attn_kernel<<<grid, THREADS, lds_bytes, stream>>>(query, attn, pos, out);
}